// ConvLayer_53137335386622
// MI455X (gfx1250) — compile-verified
//
#include <hip/hip_runtime.h>

typedef __attribute__((ext_vector_type(2))) float v2f;
typedef __attribute__((ext_vector_type(8))) float v8f;

#define D 128
#define BN_EPS 1e-5f

// ---------------- degree / normalization ----------------

__global__ void k_init_deg(float* __restrict__ deg, int n) {
  int i = blockIdx.x * blockDim.x + threadIdx.x;
  if (i < n) deg[i] = 1.0f;  // self-loop contributes 1 to every node's degree
}

__global__ void k_zero_sums(float* __restrict__ sums) {
  int t = threadIdx.x;
  if (t < 2 * D) sums[t] = 0.0f;  // [0..127] sum, [128..255] sumsq
}

__global__ void k_deg_accum(const long long* __restrict__ dst, float* __restrict__ deg, int E) {
  int i = blockIdx.x * blockDim.x + threadIdx.x;
  if (i < E) atomicAdd(&deg[(int)dst[i]], 1.0f);
}

__global__ void k_dinv(float* __restrict__ deg, int n) {
  int i = blockIdx.x * blockDim.x + threadIdx.x;
  if (i < n) deg[i] = rsqrtf(deg[i]);  // deg >= 1 always (self loops)
}

// ---------------- GEMM h = x @ W via V_WMMA_F32_16X16X4_F32 ----------------
// Block: 256 threads = 8 waves. Block covers 16 rows; wave w owns cols [16w,16w+16).
// A tile (16x128 f32) staged in LDS with padded stride 132 (bank-conflict free).

__global__ void __launch_bounds__(256) k_gemm(const float* __restrict__ x,
                                              const float* __restrict__ W,
                                              float* __restrict__ h) {
  __shared__ float As[16 * 132];
  const int t = threadIdx.x;
  const int row_base = blockIdx.x * 16;

  // cooperative stage of the 16x128 A tile (each thread: 8 contiguous floats)
  {
    const int r = t >> 4;            // 0..15
    const int c = (t & 15) * 8;      // 0,8,...,120
    const float4* src = (const float4*)&x[(size_t)(row_base + r) * D + c];
    float4 v0 = src[0];
    float4 v1 = src[1];
    *(float4*)&As[r * 132 + c] = v0;
    *(float4*)&As[r * 132 + c + 4] = v1;
  }
  __syncthreads();

  const int wave = t >> 5;
  const int lane = t & 31;
  const int half = lane >> 4;   // 0: K pair {0,1}; 1: K pair {2,3} within each 16x4 step
  const int mn = lane & 15;     // row index for A frag, col index for B/C/D frags
  const int col = wave * 16 + mn;

  // Preload all B fragments for this wave's 128x16 panel of W (row-major [D_IN][D_OUT]).
  v2f bf[32];
#pragma unroll
  for (int kt = 0; kt < 32; ++kt) {
    const int k = kt * 4 + 2 * half;
    bf[kt].x = W[(size_t)k * D + col];
    bf[kt].y = W[(size_t)(k + 1) * D + col];
  }

  v8f acc = {};
#pragma unroll
  for (int kt = 0; kt < 32; ++kt) {
    const int k = kt * 4 + 2 * half;
    v2f a;
    a.x = As[mn * 132 + k];
    a.y = As[mn * 132 + k + 1];
    // (neg_a, A, neg_b, B, c_mod, C, reuse_a, reuse_b)
    acc = __builtin_amdgcn_wmma_f32_16x16x4_f32(false, a, false, bf[kt],
                                                (short)0, acc, false, false);
  }

  // C/D layout: VGPR i -> row (i + 8*half), col = lane&15
#pragma unroll
  for (int i = 0; i < 8; ++i) {
    h[(size_t)(row_base + i + 8 * half) * D + col] = acc[i];
  }
}

// ---------------- aggregation ----------------
// out[i][c] = b[c] + dinv[i]^2 * h[i][c]   (bias + self-loop term, no atomics)

__global__ void k_agg_init(const float* __restrict__ hbuf, const float* __restrict__ dinv,
                           const float* __restrict__ bias, float* __restrict__ out, int n) {
  int idx = blockIdx.x * blockDim.x + threadIdx.x;
  if (idx >= n * 32) return;
  const int row = idx >> 5;
  const int c = (idx & 31) * 4;
  const float di = dinv[row];
  const float w = di * di;
  float4 hv = *(const float4*)&hbuf[(size_t)row * D + c];
  float4 bv = *(const float4*)&bias[c];
  float4 o;
  o.x = bv.x + w * hv.x;
  o.y = bv.y + w * hv.y;
  o.z = bv.z + w * hv.z;
  o.w = bv.w + w * hv.w;
  *(float4*)&out[(size_t)row * D + c] = o;
}

// One wave per edge: lane l handles columns 4l..4l+3. L2-resident float atomics.
__global__ void __launch_bounds__(256) k_agg_edges(const long long* __restrict__ src,
                                                   const long long* __restrict__ dst,
                                                   const float* __restrict__ hbuf,
                                                   const float* __restrict__ dinv,
                                                   float* __restrict__ out, int E) {
  const int wave = threadIdx.x >> 5;
  const int lane = threadIdx.x & 31;
  const int e = blockIdx.x * 8 + wave;
  if (e >= E) return;
  const int s = (int)src[e];
  const int d = (int)dst[e];
  const float nrm = dinv[s] * dinv[d];
  float4 hv = *(const float4*)&hbuf[(size_t)s * D + lane * 4];
  float* op = &out[(size_t)d * D + lane * 4];
  atomicAdd(op + 0, nrm * hv.x);
  atomicAdd(op + 1, nrm * hv.y);
  atomicAdd(op + 2, nrm * hv.z);
  atomicAdd(op + 3, nrm * hv.w);
}

// ---------------- BatchNorm (training, biased var) + ReLU ----------------

__global__ void __launch_bounds__(256) k_bn_stats(const float* __restrict__ out,
                                                  float* __restrict__ sums,
                                                  float* __restrict__ sumsq, int n) {
  __shared__ float ls[2 * D];
  const int t = threadIdx.x;
  ls[t] = 0.0f;
  __syncthreads();

  const int c = (t & 31) * 4;
  int row = blockIdx.x * 8 + (t >> 5);
  const int stride = gridDim.x * 8;
  float s0 = 0.f, s1 = 0.f, s2 = 0.f, s3 = 0.f;
  float q0 = 0.f, q1 = 0.f, q2 = 0.f, q3 = 0.f;
  for (; row < n; row += stride) {
    float4 v = *(const float4*)&out[(size_t)row * D + c];
    s0 += v.x; s1 += v.y; s2 += v.z; s3 += v.w;
    q0 += v.x * v.x; q1 += v.y * v.y; q2 += v.z * v.z; q3 += v.w * v.w;
  }
  atomicAdd(&ls[c + 0], s0);
  atomicAdd(&ls[c + 1], s1);
  atomicAdd(&ls[c + 2], s2);
  atomicAdd(&ls[c + 3], s3);
  atomicAdd(&ls[D + c + 0], q0);
  atomicAdd(&ls[D + c + 1], q1);
  atomicAdd(&ls[D + c + 2], q2);
  atomicAdd(&ls[D + c + 3], q3);
  __syncthreads();
  if (t < D) {
    atomicAdd(&sums[t], ls[t]);
    atomicAdd(&sumsq[t], ls[D + t]);
  }
}

__global__ void k_bn_apply(float* __restrict__ out, const float* __restrict__ sums,
                           const float* __restrict__ sumsq, const float* __restrict__ gamma,
                           const float* __restrict__ beta, int n) {
  int idx = blockIdx.x * blockDim.x + threadIdx.x;
  if (idx >= n * 32) return;
  const int row = idx >> 5;
  const int c = (idx & 31) * 4;
  const float inv_n = 1.0f / (float)n;
  float* op = &out[(size_t)row * D + c];
  float4 v = *(const float4*)op;
  float vv[4] = {v.x, v.y, v.z, v.w};
  float rr[4];
#pragma unroll
  for (int j = 0; j < 4; ++j) {
    const float mean = sums[c + j] * inv_n;
    const float var = sumsq[c + j] * inv_n - mean * mean;
    const float scl = rsqrtf(var + BN_EPS) * gamma[c + j];
    const float y = (vv[j] - mean) * scl + beta[c + j];
    rr[j] = fmaxf(y, 0.0f);
  }
  *(float4*)op = *(float4*)rr;
}

// ---------------- launcher ----------------

extern "C" void kernel_launch(void* const* d_in, const int* in_sizes, int n_in,
                              void* d_out, int out_size, void* d_ws, size_t ws_size,
                              hipStream_t stream) {
  const float* x = (const float*)d_in[0];
  const long long* ei = (const long long*)d_in[1];  // int64 edge_index [2, E]
  const float* W = (const float*)d_in[2];
  const float* b = (const float*)d_in[3];
  const float* gamma = (const float*)d_in[4];
  const float* beta = (const float*)d_in[5];
  float* out = (float*)d_out;

  const int n = in_sizes[0] / D;  // 50000 (divisible by 16)
  const int E = in_sizes[1] / 2;  // 1600000
  const long long* src = ei;
  const long long* dst = ei + E;

  float* hbuf = (float*)d_ws;                  // N*128 floats
  float* deg = hbuf + (size_t)n * D;           // N floats (deg -> dinv in place)
  float* sums = deg + n;                       // 128 floats
  float* sumsq = sums + D;                     // 128 floats

  k_init_deg<<<(n + 255) / 256, 256, 0, stream>>>(deg, n);
  k_zero_sums<<<1, 256, 0, stream>>>(sums);
  k_deg_accum<<<(E + 255) / 256, 256, 0, stream>>>(dst, deg, E);
  k_dinv<<<(n + 255) / 256, 256, 0, stream>>>(deg, n);
  k_gemm<<<n / 16, 256, 0, stream>>>(x, W, hbuf);
  k_agg_init<<<(n * 32 + 255) / 256, 256, 0, stream>>>(hbuf, deg, b, out, n);
  k_agg_edges<<<(E + 7) / 8, 256, 0, stream>>>(src, dst, hbuf, deg, out, E);
  k_bn_stats<<<1024, 256, 0, stream>>>(out, sums, sumsq, n);
  k_bn_apply<<<(n * 32 + 255) / 256, 256, 0, stream>>>(out, sums, sumsq, gamma, beta, n);
}